// PINN_5085241279067
// MI455X (gfx1250) — compile-verified
//
#include <hip/hip_runtime.h>
#include <hip/hip_bf16.h>

typedef __attribute__((ext_vector_type(2))) float v2f;
typedef __attribute__((ext_vector_type(8))) float v8f;
typedef __attribute__((ext_vector_type(4))) unsigned int u32x4;
typedef __attribute__((ext_vector_type(8))) int i32x8;
typedef __attribute__((ext_vector_type(4))) int i32x4;

#define HID            128
#define NPTS           8192
#define BLOCK          256
#define WAVES          8             // 256 threads / wave32
#define PTS_PER_WAVE   2             // 16 rows = 2 points x 8 channels
#define PTS_PER_BLOCK  16
#define NBLOCKS        (NPTS / PTS_PER_BLOCK)   // 512
#define NWAVETOT       (NBLOCKS * WAVES)        // 4096

// phase-field fracture constants
#define C_MU   80.77f
#define C_LAM  121.15f
#define C_GC   2.7e-3f
#define C_LEN  0.015f

#if __has_builtin(__builtin_amdgcn_tensor_load_to_lds) && \
    __has_builtin(__builtin_amdgcn_s_wait_tensorcnt)
#define USE_TDM 1
#else
#define USE_TDM 0
#endif

__device__ __forceinline__ float wshfl(float v, int srclane) {
    return __shfl(v, srclane, 32);
}

#if USE_TDM
// ---------------------------------------------------------------------------
// Tensor Data Mover: DMA a 64-row x 128-col f32 tile of a row-major 128x128
// matrix from global memory into LDS.  D# per CDNA5 ISA §8:
//   group0: count=1 | lds_addr | global_addr(57b) | type=2
//   group1: data_size=4B, tensor_dim0=128, tensor_dim1=128, tile_dim0=128,
//           tile_dim1=64, tensor_dim0_stride=128 (elements)
// groups 2/3 zero (2D tile).  EXEC is ignored; issued once per wave.
// Completion tracked with TENSORcnt.
// ---------------------------------------------------------------------------
__device__ __forceinline__ void tdm_load_tile(const float* gsrc, void* ldsdst) {
    const unsigned long long ga = (unsigned long long)(uintptr_t)gsrc;
    const unsigned int ldsoff = (unsigned int)(uintptr_t)ldsdst; // low 32b = LDS byte addr
    u32x4 g0;
    g0.x = 1u;                                                  // count=1, user D#
    g0.y = ldsoff;                                              // lds_addr
    g0.z = (unsigned int)ga;                                    // global_addr[31:0]
    g0.w = ((unsigned int)(ga >> 32) & 0x01FFFFFFu) | (2u << 30); // addr[56:32] | type=2
    i32x8 g1;
    g1[0] = 0x00020000;              // workgroup_mask=0, data_size=2 (4 bytes)
    g1[1] = (int)(128u << 16);       // tensor_dim0 = 128      (bits 79:48, low part)
    g1[2] = (int)(128u << 16);       // tensor_dim1 = 128      (bits 111:80, low part)
    g1[3] = (int)(128u << 16);       // tile_dim0   = 128      (bits 127:112)
    g1[4] = 64;                      // tile_dim1   = 64 rows  (bits 143:128)
    g1[5] = 128;                     // tensor_dim0_stride = 128 elements (bits 207:160)
    g1[6] = 0;
    g1[7] = 0;
    const i32x4 z4 = {0, 0, 0, 0};
#if __clang_major__ >= 23
    const i32x8 z8 = {0, 0, 0, 0, 0, 0, 0, 0};
    __builtin_amdgcn_tensor_load_to_lds(g0, g1, z4, z4, z8, 0);
#else
    __builtin_amdgcn_tensor_load_to_lds(g0, g1, z4, z4, 0);
#endif
}
#endif

// ---------------------------------------------------------------------------
// tanh jet activation, in place on the 8 state tiles.
// State layout (WMMA C/D fragment): tile mt, register r, lane l holds element
//   (h = mt*16 + r + 8*(l>>4), col = l&15), col = point*8 + channel.
// Channels: 0=value 1=d/dx1 2=d/dx2 3=d/dt 4=d2/dx1x1 5=d2/dx1x2 6=d2/dx2x2 7=pad
// ---------------------------------------------------------------------------
__device__ __forceinline__ void jet_tanh(v8f S[8], int lane, int ch) {
    const int base = lane & ~7;   // value-channel lane of this (half, point)
    #pragma unroll
    for (int mt = 0; mt < 8; ++mt) {
        #pragma unroll
        for (int r = 0; r < 8; ++r) {
            float zb  = S[mt][r];
            float z0  = wshfl(zb, base);       // pre-activation value
            float zd1 = wshfl(zb, base + 1);   // dz/dx1
            float zd2 = wshfl(zb, base + 2);   // dz/dx2
            float a = tanhf(z0);
            float s = 1.0f - a * a;
            float g = (ch == 4) ? zd1 * zd1
                    : (ch == 5) ? zd1 * zd2
                    : (ch == 6) ? zd2 * zd2 : 0.0f;
            // a_uv = s*(z_uv - 2 a z_u z_v); first derivs have g==0 -> s*zb
            float res = s * (zb - 2.0f * a * g);
            res = (ch == 0) ? a : res;
            res = (ch == 7) ? 0.0f : res;
            S[mt][r] = res;
        }
    }
}

// ---------------------------------------------------------------------------
// Accumulate one 64-row K-half of Z += W^T * S using V_WMMA_F32_16X16X4_F32.
//   A (16x4) = W^T tile from LDS; B (4x16) = jet-state columns built with
//   ds_bpermute from the C/D fragment layout.
// ---------------------------------------------------------------------------
__device__ __forceinline__ void gemm_half(
    v8f Z[8], const v8f S[8], const float* sW, int half, int lh, int lo)
{
    #pragma unroll
    for (int kq = 0; kq < 16; ++kq) {
        const int kc  = half * 16 + kq;     // K-chunk of 4 (global k = 4*kc..4*kc+3)
        const int mtS = kc >> 2;            // source state tile
        const int bS  = (kc >> 1) & 1;      // source lane-half
        const int rb  = (kc & 1) * 4;       // source register base
        const int sl  = lo + 16 * bS;       // source lane
        // element (k = 4kc + 2*lh + v, n = lo) of S; register depends on
        // DEST half, so shuffle both candidates and select.
        float u0 = wshfl(S[mtS][rb + 0], sl);
        float u1 = wshfl(S[mtS][rb + 1], sl);
        float u2 = wshfl(S[mtS][rb + 2], sl);
        float u3 = wshfl(S[mtS][rb + 3], sl);
        v2f B;
        B.x = lh ? u2 : u0;
        B.y = lh ? u3 : u1;

        const int kl = kq * 4 + 2 * lh;     // local k row in staged half
        #pragma unroll
        for (int mtA = 0; mtA < 8; ++mtA) {
            // A[m][k] = W[k][m_global], m_global = mtA*16 + lo
            v2f A;
            A.x = sW[(kl + 0) * HID + mtA * 16 + lo];
            A.y = sW[(kl + 1) * HID + mtA * 16 + lo];
            Z[mtA] = __builtin_amdgcn_wmma_f32_16x16x4_f32(
                false, A, false, B, (short)0, Z[mtA], false, false);
        }
    }
}

// ---------------------------------------------------------------------------
__global__ __launch_bounds__(BLOCK) void pinn_jet(
    const float* __restrict__ x,  const float* __restrict__ t,
    const float* __restrict__ w0, const float* __restrict__ b0,
    const float* __restrict__ w1, const float* __restrict__ b1,
    const float* __restrict__ w2, const float* __restrict__ b2,
    const float* __restrict__ w3, const float* __restrict__ b3,
    float* __restrict__ partials)
{
    __shared__ float sW[2][64 * HID];   // 2 x 32 KB double-buffered weight tiles

    const int tid  = threadIdx.x;
    const int wave = tid >> 5;
    const int lane = tid & 31;
    const int lh   = lane >> 4;
    const int lo   = lane & 15;
    const int ch   = lane & 7;
    const int p    = (lane >> 3) & 1;
    const int pid  = blockIdx.x * PTS_PER_BLOCK + wave * PTS_PER_WAVE + p;

    __builtin_prefetch(w3, 0, 1);       // global_prefetch of tail tensors
    __builtin_prefetch(b3, 0, 1);

#if USE_TDM
    // kick off the first two weight DMAs before doing any compute
    if (wave == 0) {
        tdm_load_tile(w1,            &sW[0][0]);   // stage 0: W1 rows   0..63
        tdm_load_tile(w1 + 64 * HID, &sW[1][0]);   // stage 1: W1 rows  64..127
    }
#endif

    // ---- layer 0: input jet -> Z -> tanh jet ----
    const float x1 = x[2 * pid], x2 = x[2 * pid + 1], tt = t[pid];
    v8f S[8];
    #pragma unroll
    for (int mt = 0; mt < 8; ++mt) {
        #pragma unroll
        for (int r = 0; r < 8; ++r) {
            const int h = mt * 16 + r + 8 * lh;
            const float wa = w0[h], wb = w0[HID + h], wc = w0[2 * HID + h];
            float zv = (ch == 0) ? (x1 * wa + x2 * wb + tt * wc + b0[h])
                     : (ch == 1) ? wa
                     : (ch == 2) ? wb
                     : (ch == 3) ? wc : 0.0f;   // 2nd derivs of affine input = 0
            S[mt][r] = zv;
        }
    }
    jet_tanh(S, lane, ch);

    // ---- hidden layers: 4 pipelined stages (W1 lo, W1 hi, W2 lo, W2 hi) ----
    v8f Z[8];
    #pragma unroll
    for (int mt = 0; mt < 8; ++mt) Z[mt] = v8f{0, 0, 0, 0, 0, 0, 0, 0};

    #pragma unroll
    for (int s = 0; s < 4; ++s) {
        const int buf = s & 1;
#if USE_TDM
        if (wave == 0) {
            if (s < 3) __builtin_amdgcn_s_wait_tensorcnt(1);   // stage-s DMA done
            else       __builtin_amdgcn_s_wait_tensorcnt(0);   // last stage: drain
        }
        __syncthreads();                 // stage-s tile visible to all waves
#else
        __syncthreads();
        {
            const float* src = (s < 2 ? w1 : w2) + (s & 1) * 64 * HID;
            const float4* g4 = reinterpret_cast<const float4*>(src);
            float4*       s4 = reinterpret_cast<float4*>(&sW[buf][0]);
            #pragma unroll
            for (int i = 0; i < 8; ++i) s4[tid + i * BLOCK] = g4[tid + i * BLOCK];
            __syncthreads();
        }
#endif
        gemm_half(Z, S, &sW[buf][0], s & 1, lh, lo);
        __syncthreads();                 // all waves done reading this buffer
#if USE_TDM
        if (s < 2 && wave == 0)          // refill buffer with stage s+2 tile
            tdm_load_tile(w2 + (s & 1) * 64 * HID, &sW[buf][0]);
#endif
        if (s & 1) {                     // layer complete: bias + tanh jet
            const float* b = (s == 1) ? b1 : b2;
            #pragma unroll
            for (int mt = 0; mt < 8; ++mt) {
                #pragma unroll
                for (int r = 0; r < 8; ++r) {
                    const int h = mt * 16 + r + 8 * lh;
                    S[mt][r] = Z[mt][r] + ((ch == 0) ? b[h] : 0.0f);
                }
            }
            jet_tanh(S, lane, ch);
            if (s == 1) {
                #pragma unroll
                for (int mt = 0; mt < 8; ++mt) Z[mt] = v8f{0, 0, 0, 0, 0, 0, 0, 0};
            }
        }
    }

    // ---- output layer: D = W3^T * S  (rows 0..2 valid) ----
    v8f D = {0, 0, 0, 0, 0, 0, 0, 0};
    #pragma unroll
    for (int kc = 0; kc < 32; ++kc) {
        const int mtS = kc >> 2;
        const int bS  = (kc >> 1) & 1;
        const int rb  = (kc & 1) * 4;
        const int sl  = lo + 16 * bS;
        float u0 = wshfl(S[mtS][rb + 0], sl);
        float u1 = wshfl(S[mtS][rb + 1], sl);
        float u2 = wshfl(S[mtS][rb + 2], sl);
        float u3 = wshfl(S[mtS][rb + 3], sl);
        v2f B; B.x = lh ? u2 : u0; B.y = lh ? u3 : u1;
        const int kA = kc * 4 + 2 * lh;
        v2f A;                       // A[m][k] = W3[k][m], m = lo (<3)
        A.x = (lo < 3) ? w3[(kA + 0) * 3 + lo] : 0.0f;
        A.y = (lo < 3) ? w3[(kA + 1) * 3 + lo] : 0.0f;
        D = __builtin_amdgcn_wmma_f32_16x16x4_f32(
            false, A, false, B, (short)0, D, false, false);
    }

    // ---- gather output jets; lanes 0 and 1 evaluate the physics ----
    const int myp = lane;            // point slot if lane < 2
    float s0[7], s1[7], s2[7];       // jets of sol0 (phi-pre), u1, u2
    #pragma unroll
    for (int c = 0; c < 7; ++c) {
        const int src = (myp * 8 + c) & 31;
        s0[c] = wshfl(D[0], src);
        s1[c] = wshfl(D[1], src);
        s2[c] = wshfl(D[2], src);
    }

    const float phi = expf(-(s0[0] + b3[0]));          // phi = exp(-sol0)
    // irreversibility: relu(-dphi/dt) = relu(phi * dsol0/dt)
    const float c_irr = fmaxf(phi * s0[3], 0.0f);
    // lap phi = phi * ((z1^2 - zxx) + (z2^2 - zyy))
    const float lap_phi = phi * ((s0[1] * s0[1] - s0[4]) +
                                 (s0[2] * s0[2] - s0[6]));
    // strain from first derivatives of u
    const float e11 = s1[1];
    const float e22 = s2[2];
    const float e12 = 0.5f * (s1[2] + s2[1]);
    const float tr  = e11 + e22;
    const float trp = fmaxf(tr, 0.0f);
    const float pos = 0.5f * (C_LAM + C_MU) * trp * trp;
    const float dd  = e11 - e22;
    const float psi = pos + C_MU * (0.5f * dd * dd + 2.0f * e12 * e12);
    const float pf  = C_GC * (phi / C_LEN - C_LEN * lap_phi)
                    - 2.0f * (1.0f - phi) * psi;
    // div sigma_j = mu * lap(u_j) + (mu+lam) * d_j(div u)
    const float lap_u1 = s1[4] + s1[6];
    const float lap_u2 = s2[4] + s2[6];
    const float ddiv1  = s1[4] + s2[5];
    const float ddiv2  = s1[5] + s2[6];
    const float ds1 = C_MU * lap_u1 + (C_MU + C_LAM) * ddiv1;
    const float ds2 = C_MU * lap_u2 + (C_MU + C_LAM) * ddiv2;
    const float om  = 1.0f - phi;
    const float res_stress = om * om * (fabsf(ds1) + fabsf(ds2));

    const float c0 = res_stress * res_stress;
    const float c1 = pf * pf;
    const float c2 = c_irr;

    // deterministic per-wave pair sum, lane 0 writes the wave partial
    const float t0 = wshfl(c0, 0) + wshfl(c0, 1);
    const float t1 = wshfl(c1, 0) + wshfl(c1, 1);
    const float t2 = wshfl(c2, 0) + wshfl(c2, 1);
    if (lane == 0) {
        const int w = blockIdx.x * WAVES + wave;
        partials[3 * w + 0] = t0;
        partials[3 * w + 1] = t1;
        partials[3 * w + 2] = t2;
    }
}

// ---------------------------------------------------------------------------
// Deterministic final reduction of 4096 wave partials -> 3 mean losses
// ---------------------------------------------------------------------------
__global__ __launch_bounds__(256) void pinn_reduce(
    const float* __restrict__ partials, float* __restrict__ out)
{
    __shared__ float red[256][3];
    const int tid = threadIdx.x;
    float a0 = 0.0f, a1 = 0.0f, a2 = 0.0f;
    for (int e = tid; e < NWAVETOT; e += 256) {
        a0 += partials[3 * e + 0];
        a1 += partials[3 * e + 1];
        a2 += partials[3 * e + 2];
    }
    red[tid][0] = a0; red[tid][1] = a1; red[tid][2] = a2;
    __syncthreads();
    if (tid < 3) {
        float s = 0.0f;
        for (int i = 0; i < 256; ++i) s += red[i][tid];
        out[tid] = s / (float)NPTS;
    }
}

// ---------------------------------------------------------------------------
extern "C" void kernel_launch(void* const* d_in, const int* in_sizes, int n_in,
                              void* d_out, int out_size, void* d_ws, size_t ws_size,
                              hipStream_t stream) {
    (void)in_sizes; (void)n_in; (void)out_size; (void)ws_size;
    const float* x  = (const float*)d_in[0];
    const float* t  = (const float*)d_in[1];
    const float* w0 = (const float*)d_in[2];
    const float* b0 = (const float*)d_in[3];
    const float* w1 = (const float*)d_in[4];
    const float* b1 = (const float*)d_in[5];
    const float* w2 = (const float*)d_in[6];
    const float* b2 = (const float*)d_in[7];
    const float* w3 = (const float*)d_in[8];
    const float* b3 = (const float*)d_in[9];
    float* partials = (float*)d_ws;   // 4096 * 3 floats = 48 KB
    float* out      = (float*)d_out;

    pinn_jet<<<dim3(NBLOCKS), dim3(BLOCK), 0, stream>>>(
        x, t, w0, b0, w1, b1, w2, b2, w3, b3, partials);
    pinn_reduce<<<dim3(1), dim3(256), 0, stream>>>(partials, out);
}